// GCN_31576599560549
// MI455X (gfx1250) — compile-verified
//
#include <hip/hip_runtime.h>

typedef __attribute__((ext_vector_type(2))) float v2f;
typedef __attribute__((ext_vector_type(4))) float v4f;
typedef __attribute__((ext_vector_type(8))) float v8f;

#define GCN_NODES   100000
#define GCN_EDGES   1600000
#define GCN_KDIM    128      // IN_FEATS == N_HIDDEN == 128 (fixed K for all GEMMs)
#define GCN_HID     128
#define GCN_CLS     40
#define GCN_LDA     129      // padded LDS row stride (floats) -> conflict-free

// ---------------------------------------------------------------- utilities
__global__ void gcn_zero(float* __restrict__ p, long n) {
    long i = (long)blockIdx.x * blockDim.x + threadIdx.x;
    long stride = (long)gridDim.x * blockDim.x;
    for (; i < n; i += stride) p[i] = 0.0f;
}

__global__ void gcn_degrees(const int* __restrict__ src, const int* __restrict__ dst,
                            float* __restrict__ odeg, float* __restrict__ ideg, int nE) {
    int i = blockIdx.x * blockDim.x + threadIdx.x;
    int stride = gridDim.x * blockDim.x;
    for (; i < nE; i += stride) {
        atomicAdd(&odeg[src[i]], 1.0f);
        atomicAdd(&ideg[dst[i]], 1.0f);
    }
}

__global__ void gcn_rsqrt_clamp(float* __restrict__ p, int n) {
    int i = blockIdx.x * blockDim.x + threadIdx.x;
    if (i < n) p[i] = rsqrtf(fmaxf(p[i], 1.0f));
}

// ---------------------------------------------------------------- W pre-pack
// Wf[t][k4][lane][0..1]: fragment-ready B operands for V_WMMA_F32_16X16X4_F32.
// lane16 = col-in-tile, half = lane>>4, kr = 4*k4 + 2*half; zero-pad col >= N.
__global__ void gcn_wfrag(const float* __restrict__ W, float* __restrict__ Wf,
                          int N, int nColTiles) {
    int i = blockIdx.x * blockDim.x + threadIdx.x;   // (t*32 + k4)*32 + lane
    int total = nColTiles * 32 * 32;
    if (i >= total) return;
    int lane   = i & 31;
    int k4     = (i >> 5) & 31;
    int t      = i >> 10;
    int lane16 = lane & 15;
    int half   = lane >> 4;
    int kr     = 4 * k4 + 2 * half;
    int col    = t * 16 + lane16;
    float v0 = 0.0f, v1 = 0.0f;
    if (col < N) {
        v0 = W[kr * N + col];
        v1 = W[(kr + 1) * N + col];
    }
    Wf[2 * i]     = v0;
    Wf[2 * i + 1] = v1;
}

// ---------------------------------------------------------------- WMMA GEMM
// Y[r][c] = sum_k (X[r][k] * ns[r]) * W[k][c],  K fixed = 128.
// Block: 256 threads = 8 wave32. blockIdx.x -> 32-row tile; wave -> 16-col tile.
// Each wave keeps two accumulators (rows r0..r0+15 and r0+16..r0+31) fed by one
// shared B fragment -> 2 independent WMMA chains, branch-free inner loop.
__global__ void __launch_bounds__(256)
gcn_gemm_wmma(const float* __restrict__ X, const float* __restrict__ ns,
              const float* __restrict__ Wf, float* __restrict__ Y,
              int N, int nColTiles) {
    __shared__ float Alds[32 * GCN_LDA];

    const int tid = threadIdx.x;
    const int r0  = blockIdx.x * 32;

    // Stage 32x128 A tile, pre-scaled by norm_src (shared by all 8 waves).
    #pragma unroll
    for (int idx = tid; idx < 32 * GCN_KDIM; idx += 256) {
        int m = idx >> 7;          // / 128
        int k = idx & 127;
        Alds[m * GCN_LDA + k] = X[(long)(r0 + m) * GCN_KDIM + k] * ns[r0 + m];
    }
    __syncthreads();

    const int wave   = tid >> 5;
    const int lane   = tid & 31;
    const int lane16 = lane & 15;
    const int half   = lane >> 4;
    if (wave >= nColTiles) return;             // wave-uniform: EXEC all-1s below

    // A fragments: lane16 = M row; VGPR v / lane-half h hold K = k + 2h + v.
    const float* As0 = &Alds[lane16 * GCN_LDA + 2 * half];
    const float* As1 = As0 + 16 * GCN_LDA;
    // B fragments: coalesced b64 per lane per K-step.
    const v2f* Wfp = (const v2f*)Wf + (long)wave * 32 * 32 + lane;

    v8f acc0 = {0.f, 0.f, 0.f, 0.f, 0.f, 0.f, 0.f, 0.f};
    v8f acc1 = {0.f, 0.f, 0.f, 0.f, 0.f, 0.f, 0.f, 0.f};

    #pragma unroll 8
    for (int k4 = 0; k4 < GCN_KDIM / 4; ++k4) {
        const int k = k4 * 4;
        v2f a0, a1;
        a0.x = As0[k];
        a0.y = As0[k + 1];
        a1.x = As1[k];
        a1.y = As1[k + 1];
        v2f b = Wfp[k4 * 32];
        acc0 = __builtin_amdgcn_wmma_f32_16x16x4_f32(
                   false, a0, false, b, (short)0, acc0, false, false);
        acc1 = __builtin_amdgcn_wmma_f32_16x16x4_f32(
                   false, a1, false, b, (short)0, acc1, false, false);
    }

    const int col = wave * 16 + lane16;
    if (col < N) {                              // diverges only for N=40 stores
        #pragma unroll
        for (int r = 0; r < 8; ++r) {
            int m = r + 8 * half;               // C/D: VGPR r -> rows r / r+8
            Y[(long)(r0 + m) * N + col]      = acc0[r];
            Y[(long)(r0 + 16 + m) * N + col] = acc1[r];
        }
    }
}

// ---------------------------------------------------------------- SpMM scatter
// One wave per edge; lanes sweep the feature row in float4 chunks:
// b128 gathers + 4 coalesced f32 atomic adds. F must be a multiple of 4.
__global__ void __launch_bounds__(256)
gcn_spmm_scatter(const float* __restrict__ Xe, const int* __restrict__ src,
                 const int* __restrict__ dst, float* __restrict__ Y,
                 int F, int nE) {
    int gid  = blockIdx.x * blockDim.x + threadIdx.x;
    int e    = gid >> 5;
    int lane = gid & 31;
    if (e >= nE) return;
    int s = src[e];
    int d = dst[e];
    const v4f* xr = (const v4f*)(Xe + (long)s * F);
    float*     yr = Y + (long)d * F;
    for (int c = lane; c < (F >> 2); c += 32) {
        v4f v = xr[c];
        atomicAdd(&yr[4 * c + 0], v.x);
        atomicAdd(&yr[4 * c + 1], v.y);
        atomicAdd(&yr[4 * c + 2], v.z);
        atomicAdd(&yr[4 * c + 3], v.w);
    }
}

// ---------------------------------------------------------------- epilogue
__global__ void gcn_finish(float* __restrict__ Y, const float* __restrict__ nd,
                           const float* __restrict__ bias, int F, int doRelu,
                           int nNodes) {
    long i = (long)blockIdx.x * blockDim.x + threadIdx.x;
    long total = (long)nNodes * F;
    if (i >= total) return;
    int node = (int)(i / F);
    int f    = (int)(i - (long)node * F);
    float v  = Y[i] * nd[node] + bias[f];
    if (doRelu) v = fmaxf(v, 0.0f);
    Y[i] = v;
}

// ---------------------------------------------------------------- launcher
extern "C" void kernel_launch(void* const* d_in, const int* in_sizes, int n_in,
                              void* d_out, int out_size, void* d_ws, size_t ws_size,
                              hipStream_t stream) {
    (void)in_sizes; (void)n_in; (void)out_size; (void)ws_size;

    const float* feat = (const float*)d_in[0];
    const int*   src  = (const int*)  d_in[1];
    const int*   dst  = (const int*)  d_in[2];
    const float* W0   = (const float*)d_in[3];
    const float* b0   = (const float*)d_in[4];
    const float* W1   = (const float*)d_in[5];
    const float* b1   = (const float*)d_in[6];
    const float* W2   = (const float*)d_in[7];
    const float* b2   = (const float*)d_in[8];
    float*       out  = (float*)d_out;

    const int  Nn = GCN_NODES, Ne = GCN_EDGES, H = GCN_HID, C = GCN_CLS;
    float* ws   = (float*)d_ws;
    float* ns   = ws;                              // out-deg norm  (Nn)
    float* nd   = ws + Nn;                         // in-deg norm   (Nn)
    float* bufA = ws + 2L * Nn;                    // XW            (Nn*H)
    float* bufB = bufA + (long)Nn * H;             // hidden h      (Nn*H)
    float* Wf0  = bufB + (long)Nn * H;             // 8 tiles * 2048 floats
    float* Wf1  = Wf0 + 8 * 2048;
    float* Wf2  = Wf1 + 8 * 2048;                  // 3 tiles * 2048 floats

    const int spmmBlocks = (Ne * 32) / 256;        // one wave per edge

    // ---- degree norms + fragment-packed weights
    gcn_zero<<<512, 256, 0, stream>>>(ns, 2L * Nn);
    gcn_degrees<<<2048, 256, 0, stream>>>(src, dst, ns, nd, Ne);
    gcn_rsqrt_clamp<<<(2 * Nn + 255) / 256, 256, 0, stream>>>(ns, 2 * Nn);
    gcn_wfrag<<<(8 * 1024 + 255) / 256, 256, 0, stream>>>(W0, Wf0, H, 8);
    gcn_wfrag<<<(8 * 1024 + 255) / 256, 256, 0, stream>>>(W1, Wf1, H, 8);
    gcn_wfrag<<<(3 * 1024 + 255) / 256, 256, 0, stream>>>(W2, Wf2, C, 3);

    // ---- layer 0: feat -> bufB (ReLU)
    gcn_gemm_wmma<<<Nn / 32, 256, 0, stream>>>(feat, ns, Wf0, bufA, H, 8);
    gcn_zero<<<2048, 256, 0, stream>>>(bufB, (long)Nn * H);
    gcn_spmm_scatter<<<spmmBlocks, 256, 0, stream>>>(bufA, src, dst, bufB, H, Ne);
    gcn_finish<<<((long)Nn * H + 255) / 256, 256, 0, stream>>>(bufB, nd, b0, H, 1, Nn);

    // ---- layer 1: bufB -> bufB (ReLU)
    gcn_gemm_wmma<<<Nn / 32, 256, 0, stream>>>(bufB, ns, Wf1, bufA, H, 8);
    gcn_zero<<<2048, 256, 0, stream>>>(bufB, (long)Nn * H);
    gcn_spmm_scatter<<<spmmBlocks, 256, 0, stream>>>(bufA, src, dst, bufB, H, Ne);
    gcn_finish<<<((long)Nn * H + 255) / 256, 256, 0, stream>>>(bufB, nd, b1, H, 1, Nn);

    // ---- layer 2: bufB -> d_out (no ReLU), N = 40 -> 3 col tiles
    gcn_gemm_wmma<<<Nn / 32, 256, 0, stream>>>(bufB, ns, Wf2, bufA, C, 3);
    gcn_zero<<<1024, 256, 0, stream>>>(out, (long)Nn * C);
    gcn_spmm_scatter<<<spmmBlocks, 256, 0, stream>>>(bufA, src, dst, out, C, Ne);
    gcn_finish<<<((long)Nn * C + 255) / 256, 256, 0, stream>>>(out, nd, b2, C, 0, Nn);
}